// FastPitchJIT_53566832116373
// MI455X (gfx1250) — compile-verified
//
#include <hip/hip_runtime.h>
#include <hip/hip_bf16.h>

typedef float v2f __attribute__((ext_vector_type(2)));
typedef float v4f __attribute__((ext_vector_type(4)));
typedef float v8f __attribute__((ext_vector_type(8)));

#define BATCH 32
#define TLEN  1024
#define HDIM  384
#define MEL   4096
// tiles: M over output frames (MEL/16 = 256), N over channels (HDIM/16 = 24)
#define NTILES (HDIM / 16)        // 24
#define MTILES (MEL / 16)         // 256
#define WAVES_TOTAL (BATCH * MTILES * NTILES)   // 196608
#define WAVES_PER_BLOCK 8
#define BLOCKS_FUSED (WAVES_TOTAL / WAVES_PER_BLOCK) // 24576

// ---------------------------------------------------------------------------
// Kernel 1: per-batch inclusive scan of durations (reps == durations, PACE=1)
// csum[b,t], dec_i[b] = min(csum[b,T-1], MEL), dec_f[b] = (float)dec_i[b]
// ---------------------------------------------------------------------------
__global__ __launch_bounds__(256) void scan_kernel(
    const int* __restrict__ dur,
    int* __restrict__ csum,
    int* __restrict__ dec_i,
    float* __restrict__ dec_f)
{
    __shared__ int lds[256];
    const int b   = blockIdx.x;
    const int tid = threadIdx.x;

    const int* d = dur + b * TLEN + tid * 4;
    int v0 = d[0], v1 = d[1], v2 = d[2], v3 = d[3];
    int s0 = v0, s1 = s0 + v1, s2 = s1 + v2, s3 = s2 + v3;

    lds[tid] = s3;
    __syncthreads();
    int acc = s3;
    #pragma unroll
    for (int off = 1; off < 256; off <<= 1) {
        int u = (tid >= off) ? lds[tid - off] : 0;
        __syncthreads();
        acc += u;
        lds[tid] = acc;
        __syncthreads();
    }
    const int prefix = acc - s3;   // exclusive prefix of this thread's chunk

    int* cp = csum + b * TLEN + tid * 4;
    cp[0] = prefix + s0; cp[1] = prefix + s1;
    cp[2] = prefix + s2; cp[3] = prefix + s3;

    if (tid == 255) {
        int dec = acc < MEL ? acc : MEL;
        dec_i[b] = dec;
        dec_f[b] = (float)dec;     // dec_lens appended to d_out as float
    }
}

// ---------------------------------------------------------------------------
// Kernel 2: fused conditioning + length regulation.
// For output frame m of batch b: token = upper_bound(csum[b], m), and
//   out[b,m,:] = (m < dec) ? enc[b,token,:] + s[token] @ Wc + bias : 0
// where s[t] = {p[t-1],p[t],p[t+1], e[t-1],e[t],e[t+1]} (SAME padding) and
// Wc[6,H] stacks pitch_w / energy_w. Done as 16x16 tiles with
// V_WMMA_F32_16X16X4_F32 (two ops: K=0..3, K=4..5 + zero pad), C = gathered
// enc tile + biases. One wave per tile.
// ---------------------------------------------------------------------------
__global__ __launch_bounds__(256) void fused_wmma_regulate_kernel(
    const float* __restrict__ enc,     // [B,T,H]
    const float* __restrict__ pitch,   // [B,1,T]
    const float* __restrict__ energy,  // [B,1,T]
    const float* __restrict__ pw,      // [H,1,3]
    const float* __restrict__ pb,      // [H]
    const float* __restrict__ ew,      // [H,1,3]
    const float* __restrict__ eb,      // [H]
    const int*   __restrict__ csum,    // [B,T]
    const int*   __restrict__ dec_i,   // [B]
    float* __restrict__ out)           // [B,MEL,H]
{
    const int wave = blockIdx.x * WAVES_PER_BLOCK + (threadIdx.x >> 5);
    const int lane = threadIdx.x & 31;
    const int l15  = lane & 15;
    const int halfsel = lane >> 4;     // 0: lanes 0-15, 1: lanes 16-31

    int r  = wave;
    const int b  = r / (MTILES * NTILES); r -= b * (MTILES * NTILES);
    const int mt = r / NTILES;
    const int nt = r - mt * NTILES;

    const int dec = dec_i[b];

    // --- per-lane frame index + binary search for its token (row M = l15) ---
    const int m_lane = mt * 16 + l15;
    const int* cs = csum + b * TLEN;
    int lo = 0, hi = TLEN;
    #pragma unroll
    for (int it = 0; it < 10; ++it) {
        int mid = (lo + hi) >> 1;
        mid = mid < (TLEN - 1) ? mid : (TLEN - 1);   // stay in-bounds
        int cv = cs[mid];
        bool go = (cv <= m_lane);
        lo = go ? mid + 1 : lo;
        hi = go ? hi : mid;
    }
    int tok = lo < (TLEN - 1) ? lo : (TLEN - 1);     // clamp like reference

    // --- A tiles: conv window of this row's token -------------------------
    // 32-bit A 16x4 layout: v0 = A[M=l15, K=0|2], v1 = A[M=l15, K=1|3]
    const float* pp = pitch  + (size_t)b * TLEN;
    const float* ee = energy + (size_t)b * TLEN;
    const float pm1 = (tok > 0)        ? pp[tok - 1] : 0.f;
    const float p0  = pp[tok];
    const float pp1 = (tok < TLEN - 1) ? pp[tok + 1] : 0.f;
    const float em1 = (tok > 0)        ? ee[tok - 1] : 0.f;
    const float e0  = ee[tok];
    const float ep1 = (tok < TLEN - 1) ? ee[tok + 1] : 0.f;

    v2f a1, a2;
    a1[0] = halfsel ? pp1 : pm1;   // K=2 : K=0
    a1[1] = halfsel ? em1 : p0;    // K=3 : K=1
    a2[0] = halfsel ? 0.f : e0;    // K=6(pad) : K=4
    a2[1] = halfsel ? 0.f : ep1;   // K=7(pad) : K=5

    // --- B tiles: Wc[k, n], n = nt*16 + l15 -------------------------------
    // 32-bit B 4x16 layout (mirror of A): v0 = B[K=0|2, N], v1 = B[K=1|3, N]
    const int n = nt * 16 + l15;
    v2f b1, b2;
    b1[0] = halfsel ? pw[n * 3 + 2] : pw[n * 3 + 0];
    b1[1] = halfsel ? ew[n * 3 + 0] : pw[n * 3 + 1];
    b2[0] = halfsel ? 0.f           : ew[n * 3 + 1];
    b2[1] = halfsel ? 0.f           : ew[n * 3 + 2];

    // --- C tile: gathered enc rows + biases -------------------------------
    // C/D layout: VGPR r holds row M = r + halfsel*8, column N = n.
    const float bias = pb[n] + eb[n];
    v8f c;
    #pragma unroll
    for (int rr = 0; rr < 8; ++rr) {
        const int row = halfsel * 8 + rr;              // tile row 0..15
        const int tok_r = __shfl(tok, row, 32);        // that row's token
        c[rr] = enc[((size_t)b * TLEN + tok_r) * (size_t)HDIM + n] + bias;
    }

    // --- two f32 WMMAs: D = A1*B1 + (A2*B2 + C) ---------------------------
    c = __builtin_amdgcn_wmma_f32_16x16x4_f32(false, a2, false, b2,
                                              (short)0, c, false, false);
    c = __builtin_amdgcn_wmma_f32_16x16x4_f32(false, a1, false, b1,
                                              (short)0, c, false, false);

    // --- masked store: zero rows with frame >= dec ------------------------
    float* op = out + ((size_t)b * MEL + (size_t)(mt * 16 + halfsel * 8))
                      * (size_t)HDIM + n;
    #pragma unroll
    for (int rr = 0; rr < 8; ++rr) {
        const int m_r = mt * 16 + halfsel * 8 + rr;
        op[(size_t)rr * HDIM] = (m_r < dec) ? c[rr] : 0.f;
    }
}

// ---------------------------------------------------------------------------
extern "C" void kernel_launch(void* const* d_in, const int* in_sizes, int n_in,
                              void* d_out, int out_size, void* d_ws, size_t ws_size,
                              hipStream_t stream)
{
    const float* enc    = (const float*)d_in[0];  // [B,T,H]
    const float* pitch  = (const float*)d_in[1];  // [B,1,T]
    const float* energy = (const float*)d_in[2];  // [B,1,T]
    const float* pw     = (const float*)d_in[3];  // [H,1,3]
    const float* pb     = (const float*)d_in[4];  // [H]
    const float* ew     = (const float*)d_in[5];  // [H,1,3]
    const float* eb     = (const float*)d_in[6];  // [H]
    const int*   dur    = (const int*)d_in[7];    // [B,T]

    float* out   = (float*)d_out;
    float* dec_f = out + (size_t)BATCH * MEL * HDIM;   // dec_lens tail

    // scratch: csum [B,T] int + dec_i [B] int  (~132 KB)
    int* csum  = (int*)d_ws;
    int* dec_i = csum + BATCH * TLEN;

    scan_kernel<<<dim3(BATCH), dim3(256), 0, stream>>>(dur, csum, dec_i, dec_f);

    fused_wmma_regulate_kernel<<<dim3(BLOCKS_FUSED), dim3(256), 0, stream>>>(
        enc, pitch, energy, pw, pb, ew, eb, csum, dec_i, out);
}